// EdgeCrossingLoss_15925738734018
// MI455X (gfx1250) — compile-verified
//
#include <hip/hip_runtime.h>
#include <hip/hip_bf16.h>
#include <stdint.h>

typedef float v2f __attribute__((ext_vector_type(2)));
typedef float v8f __attribute__((ext_vector_type(8)));

#define K_NEIGH 32
#define EDGE_STRIDE 12   // 9 floats of edge data padded to 12 (48B, b128-friendly)
#define NB_STAGE1 256    // stage-1 reduction blocks (1 wave each)

// ---------------------------------------------------------------------------
// Kernel A: edges[f, i] = v[faces[f, (i+1)%3]] - v[faces[f, i]]  (padded to 12)
// ---------------------------------------------------------------------------
__global__ void edges_kernel(const float* __restrict__ vertices,
                             const int* __restrict__ faces,
                             float* __restrict__ edges, int F) {
    int f = blockIdx.x * blockDim.x + threadIdx.x;
    if (f >= F) return;
    int i0 = faces[3 * f + 0];
    int i1 = faces[3 * f + 1];
    int i2 = faces[3 * f + 2];
    float v0x = vertices[3 * i0 + 0], v0y = vertices[3 * i0 + 1], v0z = vertices[3 * i0 + 2];
    float v1x = vertices[3 * i1 + 0], v1y = vertices[3 * i1 + 1], v1z = vertices[3 * i1 + 2];
    float v2x = vertices[3 * i2 + 0], v2y = vertices[3 * i2 + 1], v2z = vertices[3 * i2 + 2];
    float ev[12];
    ev[0] = v1x - v0x; ev[1]  = v1y - v0y; ev[2]  = v1z - v0z;   // e0 = v1 - v0
    ev[3] = v2x - v1x; ev[4]  = v2y - v1y; ev[5]  = v2z - v1z;   // e1 = v2 - v1
    ev[6] = v0x - v2x; ev[7]  = v0y - v2y; ev[8]  = v0z - v2z;   // e2 = v0 - v2
    ev[9] = 0.f;       ev[10] = 0.f;       ev[11] = 0.f;
    float4* out = (float4*)(edges + (size_t)f * EDGE_STRIDE);
    out[0] = make_float4(ev[0], ev[1], ev[2],  ev[3]);
    out[1] = make_float4(ev[4], ev[5], ev[6],  ev[7]);
    out[2] = make_float4(ev[8], ev[9], ev[10], ev[11]);
}

// t = n/d in [0,1] without dividing; matches IEEE semantics of the reference:
// d==0 -> t is inf/NaN -> mask false; n==0,d!=0 -> t = +/-0 -> in range.
__device__ __forceinline__ bool in_range(float n, float d) {
    if (d > 0.f) return (n >= 0.f) && (n <= d);
    if (d < 0.f) return (n <= 0.f) && (n >= d);
    return false;  // d == 0 or NaN
}

// ---------------------------------------------------------------------------
// Kernel B: one wave32 per face, lane = neighbor index k (K==32 exactly).
// partials[f] = face_probs[f] * (#crossings over 3 x 32 x 3 combos)
// ---------------------------------------------------------------------------
__global__ void crossings_kernel(const float* __restrict__ edges,
                                 const int* __restrict__ nearest,
                                 const float* __restrict__ probs,
                                 float* __restrict__ partials, int F) {
    int wave = threadIdx.x >> 5;
    int lane = threadIdx.x & 31;
    int f = blockIdx.x * (blockDim.x >> 5) + wave;
    if (f >= F) return;

    // own-face edges (uniform across the wave)
    const float* ep = edges + (size_t)f * EDGE_STRIDE;
    float ex[3][3];
#pragma unroll
    for (int i = 0; i < 3; ++i)
#pragma unroll
        for (int c = 0; c < 3; ++c) ex[i][c] = ep[i * 3 + c];

    // neighbor-face edges: 3 x b128 gather
    int nb = nearest[(size_t)f * K_NEIGH + lane];
    const float4* np4 = (const float4*)(edges + (size_t)nb * EDGE_STRIDE);
    float4 n0 = np4[0], n1 = np4[1], n2 = np4[2];
    float ne[3][3] = {{n0.x, n0.y, n0.z},
                      {n0.w, n1.x, n1.y},
                      {n1.z, n1.w, n2.x}};

    int cnt = 0;
#pragma unroll
    for (int i = 0; i < 3; ++i) {
        float ax = ex[i][0], ay = ex[i][1], az = ex[i][2];
#pragma unroll
        for (int j = 0; j < 3; ++j) {
            float bx = ne[j][0], by = ne[j][1], bz = ne[j][2];
            // cp = a x b
            float cx = ay * bz - az * by;
            float cy = az * bx - ax * bz;
            float cz = ax * by - ay * bx;
            float den  = cx * ax + cy * ay + cz * az;          // cp . e_i
            float tnum = cx * bx + cy * by + cz * bz;          // cp . ne_kj
            float unum = cx * ex[j][0] + cy * ex[j][1] + cz * ex[j][2];  // cp . e_j(own)
            if (in_range(tnum, den) && in_range(unum, den)) ++cnt;
        }
    }
    // wave32 reduction over neighbors
#pragma unroll
    for (int off = 16; off > 0; off >>= 1) cnt += __shfl_xor(cnt, off, 32);
    if (lane == 0) partials[f] = probs[f] * (float)cnt;
}

// ---------------------------------------------------------------------------
// Kernel C: stage-1 sum via V_WMMA_F32_16X16X4_F32 with an all-ones B matrix.
// Each WMMA consumes 64 f32 values (A = 16x4) and accumulates exact-f32 row
// sums into C; column N=0 of C (lanes 0 and 16) holds the 16 row sums.
// One wave per block; deterministic fixed-order accumulation.
// ---------------------------------------------------------------------------
__global__ void reduce_stage1_wmma(const float* __restrict__ p, int n,
                                   float* __restrict__ out) {
    int lane = threadIdx.x;           // blockDim.x == 32
    v8f c = {};                       // zero accumulator
    v2f ones; ones.x = 1.f; ones.y = 1.f;
    int stride = gridDim.x * 64;
    for (int base = blockIdx.x * 64; base < n; base += stride) {
        v2f a;
        int i0 = base + lane;
        int i1 = base + 32 + lane;
        a.x = (i0 < n) ? p[i0] : 0.f;
        a.y = (i1 < n) ? p[i1] : 0.f;
        c = __builtin_amdgcn_wmma_f32_16x16x4_f32(
                /*neg_a=*/false, a, /*neg_b=*/false, ones,
                /*c_mod=*/(short)0, c, /*reuse_a=*/false, /*reuse_b=*/false);
    }
    float s = c[0] + c[1] + c[2] + c[3] + c[4] + c[5] + c[6] + c[7];
    // column N==0 lives in lanes 0 (M=0..7) and 16 (M=8..15)
    float tot = __shfl(s, 0, 32) + __shfl(s, 16, 32);
    if (lane == 0) out[blockIdx.x] = tot;
}

// ---------------------------------------------------------------------------
// Kernel D: single-wave final reduction of the NB_STAGE1 partials.
// ---------------------------------------------------------------------------
__global__ void reduce_stage2(const float* __restrict__ p, int n,
                              float* __restrict__ out) {
    int lane = threadIdx.x;           // blockDim.x == 32
    float s = 0.f;
    for (int i = lane; i < n; i += 32) s += p[i];
#pragma unroll
    for (int off = 16; off > 0; off >>= 1) s += __shfl_xor(s, off, 32);
    if (lane == 0) out[0] = s;
}

extern "C" void kernel_launch(void* const* d_in, const int* in_sizes, int n_in,
                              void* d_out, int out_size, void* d_ws, size_t ws_size,
                              hipStream_t stream) {
    (void)n_in; (void)out_size; (void)ws_size;
    const float* vertices = (const float*)d_in[0];   // (V,3) f32
    const int*   faces    = (const int*)d_in[1];     // (F,3) int
    const float* probs    = (const float*)d_in[2];   // (F,)  f32
    const int*   nearest  = (const int*)d_in[3];     // (F,K) int
    int F = in_sizes[2];                             // face_probs length

    float* ws       = (float*)d_ws;
    float* edges    = ws;                            // F * 12 floats (~2.4 MB)
    float* partials = ws + (size_t)F * EDGE_STRIDE;  // F floats
    float* stage    = partials + F;                  // NB_STAGE1 floats

    edges_kernel<<<(F + 255) / 256, 256, 0, stream>>>(vertices, faces, edges, F);

    int waves_per_block = 8;                         // 256 threads = 8 waves
    int blocksB = (F + waves_per_block - 1) / waves_per_block;
    crossings_kernel<<<blocksB, 256, 0, stream>>>(edges, nearest, probs, partials, F);

    reduce_stage1_wmma<<<NB_STAGE1, 32, 0, stream>>>(partials, F, stage);
    reduce_stage2<<<1, 32, 0, stream>>>(stage, NB_STAGE1, (float*)d_out);
}